// DeepWalk_45457933861169
// MI455X (gfx1250) — compile-verified
//
#include <hip/hip_runtime.h>

typedef float v2f __attribute__((ext_vector_type(2)));
typedef float v8f __attribute__((ext_vector_type(8)));

#define D_DIM 128
#define WAVES_PER_BLK 4
#define CHUNKS 32              // 128 / K(=4)

__global__ void deepwalk_zero_out(float* out) { out[0] = 0.0f; }

// One wave32 per tile of 16 edges.
//   A = Z1[u] rows for the 16 edges (16 x 128 fp32), staged in LDS in WMMA A-layout,
//       reused across all L levels (DS pipe feeds A, VMEM pipe feeds B: no spills).
//   Per level: B = Z2[paths[v][lev]] rows (16 x 128); 32 chained V_WMMA_F32_16X16X4_F32
//   accumulate C = A * B^T (16x16); we only need diag(C) = per-edge dot products.
// C layout (ISA 7.12.2): VGPR r, lanes 0-15 -> (M=r, N=lane); lanes 16-31 -> (M=r+8, N=lane-16).
// Diagonal (i,i): i<8 at lane i / vgpr i;  i>=8 at lane i+16 / vgpr i-8.
// LDS layout: [chunk][lane] of v2f -> lane l hits banks {2l, 2l+1}: conflict-free b64 access.
__global__ void __launch_bounds__(WAVES_PER_BLK * 32)
deepwalk_hs_loss(const int* __restrict__ edges,
                 const float* __restrict__ Z1,
                 const float* __restrict__ Z2,
                 const int* __restrict__ paths,
                 const float* __restrict__ signs,
                 const float* __restrict__ mask,
                 float* __restrict__ out,
                 int E, int L, int num_tiles)
{
    __shared__ float tileA[WAVES_PER_BLK * CHUNKS * 32 * 2];   // 4 x 8KB

    const int lane = threadIdx.x & 31;
    const int wid  = threadIdx.x >> 5;
    const int tile = blockIdx.x * WAVES_PER_BLK + wid;
    if (tile >= num_tiles) return;          // whole-wave exit: EXEC stays all-ones for WMMA

    float* wlds = tileA + wid * (CHUNKS * 32 * 2);

    const int base = tile * 16;
    const int row  = lane & 15;             // edge within tile: A-row and B-column owned by lane
    const int kh   = lane >> 4;             // K sub-half: lanes 0-15 hold K={0,1}, 16-31 K={2,3}

    int e = base + row; if (e >= E) e = E - 1;
    const int u = edges[2 * e + 0];
    const int v = edges[2 * e + 1];

    // ---- stage A tile into LDS in WMMA f32 16x16x4 A-layout: 32 K-chunks of v2f ----
    {
        const float* zrow = Z1 + (long)u * D_DIM + 2 * kh;
        #pragma unroll 8
        for (int c = 0; c < CHUNKS; ++c)
            *(v2f*)(wlds + (c * 32 + lane) * 2) = *(const v2f*)(zrow + 4 * c);
    }

    // ---- diagonal-extraction role of this lane ----
    const bool act = (lane < 8) || (lane >= 24);
    const int  i   = (lane < 8) ? lane : ((lane >= 24) ? lane - 16 : 0);
    int  ei  = base + i; if (ei >= E) ei = E - 1;
    const bool live = act && ((base + i) < E);
    const int  vi   = edges[2 * ei + 1];

    float acc = 0.0f;

    for (int lev = 0; lev < L; ++lev) {
        const int p = paths[(long)v * L + lev];           // internal-node row for lane's B-column
        const float* wrow = Z2 + (long)p * D_DIM + 2 * kh;

        v8f c = {};
        #pragma unroll 8
        for (int k = 0; k < CHUNKS; ++k) {
            v2f a = *(const v2f*)(wlds + (k * 32 + lane) * 2);   // ds_load_b64, bank-conflict-free
            v2f b = *(const v2f*)(wrow + 4 * k);                 // global_load_b64 (L2-resident Z2)
            c = __builtin_amdgcn_wmma_f32_16x16x4_f32(false, a, false, b,
                                                      (short)0, c, false, false);
        }

        // diag extract: lanes 0..7 take c[lane] (i=lane); lanes 24..31 take c[lane-24] (i=lane-16)
        float xv = 0.0f;
        #pragma unroll
        for (int r = 0; r < 8; ++r) {
            bool sel = (lane == r) || (lane == r + 24);
            xv = sel ? c[r] : xv;
        }

        const float s = signs[(long)vi * L + lev];
        const float m = live ? mask[(long)vi * L + lev] : 0.0f;
        const float y = s * xv;
        // log_sigmoid(y) = min(y,0) - log1p(exp(-|y|))  (numerically stable)
        const float ls = fminf(y, 0.0f) - log1pf(__expf(-fabsf(y)));
        acc += m * ls;
    }

    // wave32 reduction; inactive lanes contributed exactly 0
    #pragma unroll
    for (int o = 16; o > 0; o >>= 1)
        acc += __shfl_xor(acc, o, 32);

    if (lane == 0)
        atomicAdd(out, -acc);
}

extern "C" void kernel_launch(void* const* d_in, const int* in_sizes, int n_in,
                              void* d_out, int out_size, void* d_ws, size_t ws_size,
                              hipStream_t stream)
{
    const int*   edges = (const int*)d_in[0];   // (E,2)
    const float* Z1    = (const float*)d_in[1]; // (N,128)
    const float* Z2    = (const float*)d_in[2]; // (N-1,128)
    const int*   paths = (const int*)d_in[3];   // (N,L)
    const float* signs = (const float*)d_in[4]; // (N,L)
    const float* maskp = (const float*)d_in[5]; // (N,L)
    float* out = (float*)d_out;

    const int E = in_sizes[0] / 2;
    const int N = in_sizes[1] / D_DIM;
    const int L = in_sizes[3] / N;

    deepwalk_zero_out<<<1, 1, 0, stream>>>(out);

    const int num_tiles = (E + 15) / 16;
    const int grid = (num_tiles + WAVES_PER_BLK - 1) / WAVES_PER_BLK;
    deepwalk_hs_loss<<<grid, WAVES_PER_BLK * 32, 0, stream>>>(
        edges, Z1, Z2, paths, signs, maskp, out, E, L, num_tiles);
}